// SemGCNLayer_16192026706179
// MI455X (gfx1250) — compile-verified
//
#include <hip/hip_runtime.h>

#define N_NODES 50000
#define N_EDGES 800000
#define IN_CH   128
#define OUT_CH  256
#define LN_EPS  1e-5f

typedef __attribute__((ext_vector_type(2))) float v2f;
typedef __attribute__((ext_vector_type(8))) float v8f;

// ---------------------------------------------------------------------------
// 1) agg <- 0 ; deg <- 1 (self loop)
// ---------------------------------------------------------------------------
__global__ void gcn_init_kernel(float* __restrict__ agg, float* __restrict__ deg) {
    const int t = blockIdx.x * blockDim.x + threadIdx.x;   // N_NODES*OUT_CH threads
    agg[t] = 0.0f;
    if (t < N_NODES) deg[t] = 1.0f;
}

// ---------------------------------------------------------------------------
// 2) degree accumulation over target nodes (edge_index is int64)
// ---------------------------------------------------------------------------
__global__ void gcn_deg_kernel(const long long* __restrict__ col,
                               float* __restrict__ deg) {
    const int e = blockIdx.x * blockDim.x + threadIdx.x;
    if (e < N_EDGES) atomicAdd(&deg[(int)col[e]], 1.0f);
}

// ---------------------------------------------------------------------------
// 3) dinv = rsqrt(deg)   (deg >= 1 due to self loop)
// ---------------------------------------------------------------------------
__global__ void gcn_rsqrt_kernel(float* __restrict__ deg) {
    const int t = blockIdx.x * blockDim.x + threadIdx.x;
    if (t < N_NODES) deg[t] = rsqrtf(deg[t]);
}

// ---------------------------------------------------------------------------
// 4) Dual GEMM via fp32 WMMA (V_WMMA_F32_16X16X4_F32):
//    H   = X @ W_gcn            -> workspace
//    OUT = X @ W_res + b_res    -> d_out (residual branch)
//    One wave computes one 16x16 tile of both outputs, sharing the A fragment.
//    A lane layout : a[v] = A[M = lane&15][K = v + 2*(lane>>4)]
//    B lane layout : b[v] = B[K = v + 2*(lane>>4)][N = lane&15]
//    D lane layout : d[v] = D[M = v + 8*(lane>>4)][N = lane&15]
// ---------------------------------------------------------------------------
__global__ void gcn_gemm_kernel(const float* __restrict__ X,
                                const float* __restrict__ Wg,
                                const float* __restrict__ Wr,
                                const float* __restrict__ b_res,
                                float* __restrict__ H,
                                float* __restrict__ OUT) {
    const int wave   = threadIdx.x >> 5;          // 8 waves per block
    const int lane   = threadIdx.x & 31;
    const int lane16 = lane & 15;
    const int half   = lane >> 4;

    const int m0 = blockIdx.x * 16;                       // 3125 M-tiles
    const int n0 = (blockIdx.y * 8 + wave) * 16;          // 16 N-tiles across gridDim.y=2

    const float* xrow = X + (size_t)(m0 + lane16) * IN_CH;

    v8f acc_h = {};
    v8f acc_r = {};

    for (int k0 = 0; k0 < IN_CH; k0 += 4) {
        const int kb = k0 + 2 * half;                 // K index of this lane-half
        v2f a = *(const v2f*)(xrow + kb);             // 8B-aligned (kb even)

        const float* wg = Wg + (size_t)kb * OUT_CH + n0 + lane16;
        const float* wr = Wr + (size_t)kb * OUT_CH + n0 + lane16;
        v2f bg; bg.x = wg[0]; bg.y = wg[OUT_CH];
        v2f br; br.x = wr[0]; br.y = wr[OUT_CH];

        acc_h = __builtin_amdgcn_wmma_f32_16x16x4_f32(
            false, a, false, bg, (short)0, acc_h, false, false);
        acc_r = __builtin_amdgcn_wmma_f32_16x16x4_f32(
            false, a, false, br, (short)0, acc_r, false, false);
    }

    const float bias = b_res[n0 + lane16];
    #pragma unroll
    for (int v = 0; v < 8; ++v) {
        const int m = m0 + v + 8 * half;
        const size_t idx = (size_t)m * OUT_CH + n0 + lane16;
        H[idx]   = acc_h[v];
        OUT[idx] = acc_r[v] + bias;
    }
}

// ---------------------------------------------------------------------------
// 5) Scatter: one 256-thread block per (edge | self-loop node), thread = channel.
//    agg[col] += h[row] * dinv[row]*dinv[col]   (coalesced atomics per row)
// ---------------------------------------------------------------------------
__global__ void gcn_scatter_kernel(const long long* __restrict__ row,
                                   const long long* __restrict__ col,
                                   const float* __restrict__ dinv,
                                   const float* __restrict__ H,
                                   float* __restrict__ agg) {
    const int b  = blockIdx.x;          // [0, E+N)
    const int ch = threadIdx.x;         // [0, 256)
    int r, c;
    float w;
    if (b < N_EDGES) {
        r = (int)row[b];
        c = (int)col[b];
        w = dinv[r] * dinv[c];
    } else {
        r = c = b - N_EDGES;            // self loop
        const float d = dinv[r];
        w = d * d;
    }
    atomicAdd(&agg[(size_t)c * OUT_CH + ch], H[(size_t)r * OUT_CH + ch] * w);
}

// ---------------------------------------------------------------------------
// 6) LayerNorm + ReLU + residual. One wave32 per node, 8 channels per lane.
// ---------------------------------------------------------------------------
__global__ void gcn_ln_kernel(const float* __restrict__ agg,
                              const float* __restrict__ b_gcn,
                              const float* __restrict__ gamma,
                              const float* __restrict__ beta,
                              float* __restrict__ out) {
    const int wave = threadIdx.x >> 5;            // 8 nodes per block
    const int lane = threadIdx.x & 31;
    const int n    = blockIdx.x * 8 + wave;
    if (n >= N_NODES) return;

    const float* a = agg + (size_t)n * OUT_CH;
    float vals[8];
    float s = 0.0f;
    #pragma unroll
    for (int j = 0; j < 8; ++j) {
        const int ch = lane + 32 * j;
        vals[j] = a[ch] + b_gcn[ch];
        s += vals[j];
    }
    #pragma unroll
    for (int off = 16; off > 0; off >>= 1) s += __shfl_xor(s, off, 32);
    const float mu = s * (1.0f / OUT_CH);

    float vs = 0.0f;
    #pragma unroll
    for (int j = 0; j < 8; ++j) { const float d = vals[j] - mu; vs += d * d; }
    #pragma unroll
    for (int off = 16; off > 0; off >>= 1) vs += __shfl_xor(vs, off, 32);
    const float rinv = rsqrtf(vs * (1.0f / OUT_CH) + LN_EPS);

    float* o = out + (size_t)n * OUT_CH;
    #pragma unroll
    for (int j = 0; j < 8; ++j) {
        const int ch = lane + 32 * j;
        const float ln = (vals[j] - mu) * rinv * gamma[ch] + beta[ch];
        o[ch] = fmaxf(ln, 0.0f) + o[ch];   // ReLU + residual (res already in out)
    }
}

// ---------------------------------------------------------------------------
// Launch. Workspace layout (floats):
//   [0, 50176)                       deg / dinv (padded)
//   [50176, 50176 + N*256)           H = x @ W_gcn
//   [..,    .. + N*256)              agg (scatter target)
// Total ~102.6 MB.
// ---------------------------------------------------------------------------
extern "C" void kernel_launch(void* const* d_in, const int* in_sizes, int n_in,
                              void* d_out, int out_size, void* d_ws, size_t ws_size,
                              hipStream_t stream) {
    (void)in_sizes; (void)n_in; (void)out_size; (void)ws_size;

    const float*     x     = (const float*)d_in[0];
    const long long* ei    = (const long long*)d_in[1];   // int64 [2, E]
    const float*     Wg    = (const float*)d_in[2];
    const float*     bg    = (const float*)d_in[3];
    const float*     gamma = (const float*)d_in[4];
    const float*     beta  = (const float*)d_in[5];
    const float*     Wr    = (const float*)d_in[6];
    const float*     br    = (const float*)d_in[7];
    float*           out   = (float*)d_out;

    float* deg = (float*)d_ws;                            // doubles as dinv
    float* H   = deg + 50176;
    float* agg = H + (size_t)N_NODES * OUT_CH;

    const long long* row = ei;             // edge_index[0]
    const long long* col = ei + N_EDGES;   // edge_index[1]

    // 1) zero agg, deg <- 1
    gcn_init_kernel<<<(N_NODES * OUT_CH) / 256, 256, 0, stream>>>(agg, deg);
    // 2) degree accumulation
    gcn_deg_kernel<<<(N_EDGES + 255) / 256, 256, 0, stream>>>(col, deg);
    // 3) dinv = rsqrt(deg)
    gcn_rsqrt_kernel<<<(N_NODES + 255) / 256, 256, 0, stream>>>(deg);
    // 4) WMMA dual GEMM: H -> ws, residual -> out
    gcn_gemm_kernel<<<dim3(N_NODES / 16, 2), 256, 0, stream>>>(x, Wg, Wr, br, H, out);
    // 5) gather/scale/scatter-add over edges + self loops
    gcn_scatter_kernel<<<N_EDGES + N_NODES, 256, 0, stream>>>(row, col, deg, H, agg);
    // 6) LayerNorm + ReLU + residual
    gcn_ln_kernel<<<N_NODES / 8, 256, 0, stream>>>(agg, bg, gamma, beta, out);
}